// HawkesLayer_60730837565986
// MI455X (gfx1250) — compile-verified
//
#include <hip/hip_runtime.h>

typedef __attribute__((ext_vector_type(2))) float        v2f;
typedef __attribute__((ext_vector_type(8))) float        v8f;
typedef __attribute__((ext_vector_type(4))) unsigned int v4u;
typedef __attribute__((ext_vector_type(8))) int          v8i;
typedef __attribute__((ext_vector_type(4))) int          v4i;

#define NB_SEQ   64
#define NB_EVENT 2048
#define NB_TYPE  64
#define NB_FEAT  8
#define PRED_LEN 256
#define TOTAL    (NB_EVENT + PRED_LEN)
#define ALDS_PITCH 65   // 64 + 1 pad dword per row: kills 16-way bank conflicts

// ---------------------------------------------------------------------------
// Kernel 1: gather training events sequences[sid, :, :, 0] -> out rows [0,2048)
// ---------------------------------------------------------------------------
__global__ void hawkes_copy_kernel(const int* __restrict__ seq_id,
                                   const float* __restrict__ sequences,
                                   float* __restrict__ out) {
    const int sid = seq_id[0];
    const int idx = blockIdx.x * blockDim.x + threadIdx.x;   // 0..131071
    if (idx < NB_EVENT * NB_TYPE) {
        const int e = idx >> 6;
        const int k = idx & 63;
        out[idx] = sequences[(((size_t)sid * NB_EVENT + e) * NB_TYPE + k) * NB_FEAT];
    }
}

// ---------------------------------------------------------------------------
// Kernel 2: single-WGP sequential rollout.
// 128 threads = 4 wave32s; wave m owns M-tile m of the 64x64 matvec (WMMA,
// A-tiles resident in VGPRs). alpha streamed into padded LDS by the TDM,
// overlapped with the 4-chunk parallel bootstrap scan of S.
// ---------------------------------------------------------------------------
__global__ __launch_bounds__(128)
void hawkes_predict_kernel(const int* __restrict__ seq_id,
                           const float* __restrict__ spontaneous,
                           const float* __restrict__ Theta,
                           const float* __restrict__ W,
                           const float* __restrict__ Alpha,
                           float* __restrict__ out) {
    __shared__ float lds_alpha[NB_TYPE * ALDS_PITCH];
    __shared__ float lds_eff[NB_TYPE];
    __shared__ float lds_y[NB_TYPE];
    __shared__ float lds_part[3 * NB_TYPE];   // chunk partials p1, p2, p3

    const int tid  = threadIdx.x;
    const int lane = tid & 31;
    const int wave = tid >> 5;
    const int k    = tid & 63;                // type index
    const int grp  = tid >> 6;                // scan chunk group (0 or 1)
    const int sid  = seq_id[0];

    const float* srcA = Alpha + (size_t)sid * NB_TYPE * NB_TYPE;

    // ---- stage alpha[sid] (64x64 f32) into LDS with +1-dword row padding ----
#if __has_builtin(__builtin_amdgcn_tensor_load_to_lds)
    // Tensor Data Mover: 2D tile load with pad_enable (64 dwords -> +1 dword),
    // issued by wave 0 only; overlaps with the bootstrap scan below.
    if (wave == 0) {
        const unsigned long long gaddr = (unsigned long long)(const void*)srcA;
        const unsigned int ldsaddr = (unsigned int)(unsigned long long)(void*)lds_alpha;
        v4u g0;
        g0[0] = 1u;                                    // count=1, user descriptor
        g0[1] = ldsaddr;                               // lds_addr (byte offset)
        g0[2] = (unsigned int)(gaddr & 0xFFFFFFFFull); // global_addr[31:0]
        g0[3] = (unsigned int)((gaddr >> 32) & 0x1FFFFFFull) | (2u << 30); // addr[56:32] | type=2
        v8i g1;
        g1[0] = (int)((2u << 16)      // data_size = 4B
                    | (1u << 20)      // pad_enable
                    | (5u << 22)      // pad_interval: 64 dwords
                    | (0u << 25));    // pad_amount: 1 dword
        g1[1] = (int)(64u << 16);     // tensor_dim0 = 64 (bits 79:48, low half)
        g1[2] = (int)(64u << 16);     // dim0 hi=0 | tensor_dim1 = 64 (bits 111:80, low half)
        g1[3] = (int)(64u << 16);     // dim1 hi=0 | tile_dim0 = 64
        g1[4] = (int)(64u);           // tile_dim1 = 64 | tile_dim2 = 0
        g1[5] = (int)(64u);           // tensor_dim0_stride = 64 (low 32)
        g1[6] = 0;                    // stride hi | tensor_dim1_stride lo
        g1[7] = 0;
        v4i g2 = {0, 0, 0, 0};
        v4i g3 = {0, 0, 0, 0};
        v8i g4 = {0, 0, 0, 0, 0, 0, 0, 0};   // 6-arg toolchain: extra group, unused
        __builtin_amdgcn_tensor_load_to_lds(g0, g1, g2, g3, g4, 0);
    }
#else
    for (int i = tid; i < NB_TYPE * NB_TYPE; i += 128)
        lds_alpha[(i >> 6) * ALDS_PITCH + (i & 63)] = srcA[i];
#endif

    // ---- per-type constants (uniform code path; only tid<64 results used) ----
    const float w      = W[(size_t)sid * NB_TYPE + k];
    const float theta  = Theta[(size_t)sid * NB_TYPE + k];
    const float spont  = spontaneous[(size_t)sid * NB_TYPE + k];
    const float ew     = expf(-w);                        // per-step decay of S
    const float etheta = expf(-theta);
    const float c0     = spont / theta * (1.0f - etheta); // term1 = c0 * e^{-theta t}
    const float c1     = 1.0f - ew;                       // effect = c1 * S
    const float invw   = 1.0f / w;
    float       f      = expf(-theta * (float)NB_EVENT);  // e^{-theta t}, t = 2048

    // ---- bootstrap S(2048): 4 chunks of 512 rows, 2 independent chains/thread.
    // Chain a: chunk grp, chain b: chunk grp+2. Combine: Horner in d=e^{-512w}.
    float Sa = 0.0f, Sb = 0.0f;
    {
        const float* rowA = out + ((size_t)grp * 512) * NB_TYPE + k;
        const float* rowB = out + ((size_t)(grp + 2) * 512) * NB_TYPE + k;
        for (int tao = 0; tao < 512; tao += 8) {
            __builtin_prefetch(rowA + (size_t)(tao + 16) * NB_TYPE, 0, 1);
            __builtin_prefetch(rowB + (size_t)(tao + 16) * NB_TYPE, 0, 1);
            float ra[8], rb[8];
#pragma unroll
            for (int j = 0; j < 8; ++j) {
                ra[j] = rowA[(size_t)(tao + j) * NB_TYPE];
                rb[j] = rowB[(size_t)(tao + j) * NB_TYPE];
            }
#pragma unroll
            for (int j = 0; j < 8; ++j) {   // two independent serial chains (2x ILP)
                Sa = (Sa + ra[j]) * ew;
                Sb = (Sb + rb[j]) * ew;
            }
        }
    }
    if (tid < 64) {
        lds_part[64 + k] = Sb;            // p2
    } else {
        lds_part[k]       = Sa;           // p1
        lds_part[128 + k] = Sb;           // p3
    }

#if __has_builtin(__builtin_amdgcn_tensor_load_to_lds)
    if (wave == 0) __builtin_amdgcn_s_wait_tensorcnt(0);  // alpha tile landed
#endif
    __syncthreads();

    // S(2048) = ((p0*d + p1)*d + p2)*d + p3,  d = e^{-512 w}
    const float d512 = expf(-w * 512.0f);
    float S_run = ((Sa * d512 + lds_part[k]) * d512 + lds_part[64 + k]) * d512
                + lds_part[128 + k];      // meaningful for tid<64 (p0 == Sa there)

    // ---- hoist loop-invariant A-tiles (alpha) into VGPRs: 16 tiles x 2 VGPRs ----
    const int mrow = (wave << 4) + (lane & 15);  // this lane's alpha row
    const int kb   = (lane < 16) ? 0 : 2;        // K sub-pair per WMMA A/B layout
    v2f areg[16];
#pragma unroll
    for (int kt = 0; kt < 16; ++kt) {
        areg[kt].x = lds_alpha[mrow * ALDS_PITCH + (kt << 2) + kb];
        areg[kt].y = lds_alpha[mrow * ALDS_PITCH + (kt << 2) + kb + 1];
    }

    if (tid < 64) lds_eff[k] = S_run * c1;       // effect for first step
    __syncthreads();

    // ---- 256 sequential prediction steps: 2 barriers/step, A in registers ----
    for (int t = NB_EVENT; t < TOTAL; ++t) {
        // y[64] = alpha(64x64) @ eff(64) via V_WMMA_F32_16X16X4_F32,
        // 4 accumulator chains (4 dependent WMMAs each) to hide XDL latency.
        v2f breg[16];
#pragma unroll
        for (int kt = 0; kt < 16; ++kt) {
            breg[kt].x = lds_eff[(kt << 2) + kb];
            breg[kt].y = lds_eff[(kt << 2) + kb + 1];
        }
        v8f acc0 = {}, acc1 = {}, acc2 = {}, acc3 = {};
#pragma unroll
        for (int kt = 0; kt < 16; kt += 4) {
            acc0 = __builtin_amdgcn_wmma_f32_16x16x4_f32(
                false, areg[kt + 0], false, breg[kt + 0], (short)0, acc0, false, false);
            acc1 = __builtin_amdgcn_wmma_f32_16x16x4_f32(
                false, areg[kt + 1], false, breg[kt + 1], (short)0, acc1, false, false);
            acc2 = __builtin_amdgcn_wmma_f32_16x16x4_f32(
                false, areg[kt + 2], false, breg[kt + 2], (short)0, acc2, false, false);
            acc3 = __builtin_amdgcn_wmma_f32_16x16x4_f32(
                false, areg[kt + 3], false, breg[kt + 3], (short)0, acc3, false, false);
        }
        v8f acc = (acc0 + acc1) + (acc2 + acc3);

        // D layout (32-bit C/D 16x16): VGPR v -> M=v (lanes 0-15), M=v+8 (16-31).
        // All N columns identical (B columns replicated) -> lanes 0 and 16 export.
        if (lane == 0) {
#pragma unroll
            for (int v = 0; v < 8; ++v) lds_y[(wave << 4) + v] = acc[v];
        } else if (lane == 16) {
#pragma unroll
            for (int v = 0; v < 8; ++v) lds_y[(wave << 4) + 8 + v] = acc[v];
        }
        __syncthreads();

        if (tid < 64) {
            const float val = c0 * f + lds_y[k] * invw;  // term1 + alpha@effect/w
            out[(size_t)t * NB_TYPE + k] = val;
            S_run = (S_run + val) * ew;                  // advance exp-kernel state
            f *= etheta;                                 // advance e^{-theta t}
            lds_eff[k] = S_run * c1;                     // effect for next step
        }
        __syncthreads();
    }
}

// ---------------------------------------------------------------------------
extern "C" void kernel_launch(void* const* d_in, const int* in_sizes, int n_in,
                              void* d_out, int out_size, void* d_ws, size_t ws_size,
                              hipStream_t stream) {
    (void)in_sizes; (void)n_in; (void)out_size; (void)d_ws; (void)ws_size;
    const int*   seq_id      = (const int*)d_in[0];
    const float* sequences   = (const float*)d_in[1];
    const float* spontaneous = (const float*)d_in[2];
    const float* Theta       = (const float*)d_in[3];
    const float* W           = (const float*)d_in[4];
    const float* Alpha       = (const float*)d_in[5];
    float*       out         = (float*)d_out;

    const int ncopy  = NB_EVENT * NB_TYPE;
    const int blocks = (ncopy + 255) / 256;
    hawkes_copy_kernel<<<blocks, 256, 0, stream>>>(seq_id, sequences, out);
    hawkes_predict_kernel<<<1, 128, 0, stream>>>(seq_id, spontaneous, Theta, W,
                                                 Alpha, out);
}